// ChamferDistance_65352222376121
// MI455X (gfx1250) — compile-verified
//
#include <hip/hip_runtime.h>
#include <math.h>

typedef float v2f __attribute__((ext_vector_type(2)));
typedef float v8f __attribute__((ext_vector_type(8)));

#define CHAMFER_EPS 1e-10f

// One wave32 per 16-row tile of X; sweeps all columns of Y in 16-wide tiles.
// dist^2 tile computed by a single V_WMMA_F32_16X16X4_F32:
//   A[m][*] = (-2x0, -2x1, -2x2, |x|^2+eps), B[*][n] = (y0, y1, y2, 1), C = |y|^2
// Row-min of dist^2 kept in registers; sqrt applied only to the 16 final mins.
__global__ __launch_bounds__(256) void chamfer_rowmin_kernel(
    const float* __restrict__ X,   // (B, NX, 3)
    const float* __restrict__ Y,   // (B, NY, 3)
    float* __restrict__ outmin,    // (B, NX): min_j dist(x_i, y_j)
    int B, int NX, int NY)
{
    const int lane  = threadIdx.x & 31;
    const int wib   = threadIdx.x >> 5;             // wave in block (0..7)
    const int gwave = blockIdx.x * 8 + wib;
    const int tilesPerBatch = NX >> 4;
    const int b    = gwave / tilesPerBatch;
    if (b >= B) return;                              // uniform per-wave branch
    const int tile = gwave - b * tilesPerBatch;
    const int base_n = tile << 4;
    const int col = lane & 15;
    const bool hi = lane >= 16;

    // ---- A fragment (16x4 f32 layout: lanes 0-15 hold K=0,1; 16-31 hold K=2,3)
    const float* xp = X + ((size_t)b * NX + base_n + col) * 3;
    const float x0 = xp[0], x1 = xp[1], x2 = xp[2];
    const float xx = x0 * x0 + x1 * x1 + x2 * x2 + CHAMFER_EPS;
    v2f afrag;
    afrag.x = hi ? (-2.0f * x2) : (-2.0f * x0);      // K=2 | K=0
    afrag.y = hi ? xx           : (-2.0f * x1);      // K=3 | K=1

    v8f rmin;
#pragma unroll
    for (int k = 0; k < 8; ++k) rmin[k] = __builtin_inff();

    const float* ybase = Y + (size_t)b * NY * 3;
    for (int m = 0; m < NY; m += 16) {
        const float* yp = ybase + (size_t)(m + col) * 3;
        const float y0 = yp[0], y1 = yp[1], y2 = yp[2];
        const float yy = y0 * y0 + y1 * y1 + y2 * y2;

        v2f bfrag;                                   // B is 4x16 (K x N)
        bfrag.x = hi ? y2   : y0;                    // K=2 | K=0
        bfrag.y = hi ? 1.0f : y1;                    // K=3 | K=1

        v8f cfrag;                                   // C[m][n] = |y_n|^2 (per-lane col)
#pragma unroll
        for (int k = 0; k < 8; ++k) cfrag[k] = yy;

        // D = A x B + C = dist^2 tile (rows base_n..base_n+15, cols m..m+15)
        v8f d = __builtin_amdgcn_wmma_f32_16x16x4_f32(
            false, afrag, false, bfrag, (short)0, cfrag, false, false);

#pragma unroll
        for (int k = 0; k < 8; ++k) rmin[k] = fminf(rmin[k], d[k]);
    }

    // Min across the 16 columns held by each lane half (xor 1,2,4,8 stays in-half).
#pragma unroll
    for (int off = 1; off < 16; off <<= 1) {
#pragma unroll
        for (int k = 0; k < 8; ++k)
            rmin[k] = fminf(rmin[k], __shfl_xor(rmin[k], off, 32));
    }

    // C/D layout: VGPR k -> row base_n+k (lanes 0-15) / base_n+8+k (lanes 16-31).
    if (lane == 0 || lane == 16) {
        float* op = outmin + (size_t)b * NX + base_n + (hi ? 8 : 0);
#pragma unroll
        for (int k = 0; k < 8; ++k) op[k] = sqrtf(rmin[k]);
    }
}

// Single block: per-batch means of both min arrays, max, sum over batches.
__global__ __launch_bounds__(256) void chamfer_reduce_kernel(
    const float* __restrict__ d1,  // (B, N)
    const float* __restrict__ d2,  // (B, M)
    float* __restrict__ out, int B, int N, int M)
{
    __shared__ float sdata[256];
    const int tid = threadIdx.x;
    float total = 0.0f;
    for (int b = 0; b < B; ++b) {
        float s1 = 0.0f;
        for (int i = tid; i < N; i += 256) s1 += d1[(size_t)b * N + i];
        sdata[tid] = s1;
        __syncthreads();
        for (int s = 128; s > 0; s >>= 1) {
            if (tid < s) sdata[tid] += sdata[tid + s];
            __syncthreads();
        }
        const float mean1 = sdata[0] / (float)N;
        __syncthreads();

        float s2 = 0.0f;
        for (int i = tid; i < M; i += 256) s2 += d2[(size_t)b * M + i];
        sdata[tid] = s2;
        __syncthreads();
        for (int s = 128; s > 0; s >>= 1) {
            if (tid < s) sdata[tid] += sdata[tid + s];
            __syncthreads();
        }
        const float mean2 = sdata[0] / (float)M;
        __syncthreads();

        total += fmaxf(mean1, mean2);
    }
    if (tid == 0) out[0] = total;
}

extern "C" void kernel_launch(void* const* d_in, const int* in_sizes, int n_in,
                              void* d_out, int out_size, void* d_ws, size_t ws_size,
                              hipStream_t stream)
{
    const float* x = (const float*)d_in[0];   // (B, N, 3) fp32
    const float* y = (const float*)d_in[1];   // (B, M, 3) fp32
    const int B = 8;
    const int N = in_sizes[0] / (B * 3);      // 8192
    const int M = in_sizes[1] / (B * 3);      // 8192

    float* dist1 = (float*)d_ws;              // (B, N)
    float* dist2 = dist1 + (size_t)B * N;     // (B, M)

    const int tiles1 = B * (N >> 4);          // waves for pass 1
    const int tiles2 = B * (M >> 4);          // waves for pass 2

    // Pass 1: for each x, nearest y.  Pass 2: roles swapped -> pure row-mins, no atomics.
    chamfer_rowmin_kernel<<<tiles1 / 8, 256, 0, stream>>>(x, y, dist1, B, N, M);
    chamfer_rowmin_kernel<<<tiles2 / 8, 256, 0, stream>>>(y, x, dist2, B, M, N);
    chamfer_reduce_kernel<<<1, 256, 0, stream>>>(dist1, dist2, (float*)d_out, B, N, M);
}